// Multi_head_89163521065038
// MI455X (gfx1250) — compile-verified
//
#include <hip/hip_runtime.h>
#include <hip/hip_bf16.h>

// ---------------------------------------------------------------------------
// Mixture-of-5-experts tiny MLP, f32, N=500000 samples.
// Strategy: bucket samples by expert (t determines expert), compact indices,
// then per-expert dense GEMM with V_WMMA_F32_16X16X4_F32 (full f32 precision).
// ---------------------------------------------------------------------------

#define NSAMP 500000
#define INDIM 50
#define HIDD  50
#define NEXP  5
#define ROWW  (1 + INDIM)   // 51 floats per x row: [t, feat0..feat49]

typedef __attribute__((ext_vector_type(2))) float v2f;
typedef __attribute__((ext_vector_type(8))) float v8f;

__device__ __forceinline__ int bucket_of(float t) {
  int e = (int)(t * 5.0f);
  if (e > 4) e = 4;
  if (e < 0) e = 0;
  return e;
}

// ---- pass 0: zero counters -------------------------------------------------
__global__ void k_init(int* __restrict__ cnt) {
  if (threadIdx.x < 8) cnt[threadIdx.x] = 0;
}

// ---- pass 1: per-expert counts (LDS-aggregated) ---------------------------
__global__ void k_count(const float* __restrict__ x, int* __restrict__ cnt, int n) {
  __shared__ int s_cnt[NEXP];
  if (threadIdx.x < NEXP) s_cnt[threadIdx.x] = 0;
  __syncthreads();
  int i = blockIdx.x * blockDim.x + threadIdx.x;
  const int stride = gridDim.x * blockDim.x;
  for (; i < n; i += stride) {
    float t = x[(size_t)i * ROWW];
    atomicAdd(&s_cnt[bucket_of(t)], 1);
  }
  __syncthreads();
  if (threadIdx.x < NEXP) atomicAdd(&cnt[threadIdx.x], s_cnt[threadIdx.x]);
}

// ---- pass 2: exclusive prefix -> offsets + scatter cursors ----------------
__global__ void k_prefix(const int* __restrict__ cnt, int* __restrict__ off,
                         int* __restrict__ cur) {
  if (threadIdx.x == 0) {
    int acc = 0;
    for (int e = 0; e < NEXP; ++e) { off[e] = acc; cur[e] = acc; acc += cnt[e]; }
  }
}

// ---- pass 3: scatter sample indices, compacted per expert -----------------
__global__ void k_scatter(const float* __restrict__ x, int* __restrict__ cur,
                          int* __restrict__ perm, int n) {
  __shared__ int s_cnt[NEXP];
  __shared__ int s_base[NEXP];
  if (threadIdx.x < NEXP) s_cnt[threadIdx.x] = 0;
  __syncthreads();
  const int i = blockIdx.x * blockDim.x + threadIdx.x;
  int e = 0, local = 0;
  bool ok = (i < n);
  if (ok) {
    float t = x[(size_t)i * ROWW];
    e = bucket_of(t);
    local = atomicAdd(&s_cnt[e], 1);
  }
  __syncthreads();
  if (threadIdx.x < NEXP)
    s_base[threadIdx.x] = atomicAdd(&cur[threadIdx.x], s_cnt[threadIdx.x]);
  __syncthreads();
  if (ok) perm[s_base[e] + local] = i;
}

// ---- main kernel: per-expert WMMA GEMM ------------------------------------
// grid = (GX, NEXP); block = 256 (8 wave32). Each wave owns 16-sample tiles.
// A (16x4 f32/step): lane L holds rows M=L%16, K pair at k0 + 2*(L/16).
// B (4x16 f32/step): lane L holds cols N=L%16, K rows  k0 + 2*(L/16), +1.
// C/D (16x16 f32):   VGPR r, lanes0-15 -> M=r, lanes16-31 -> M=r+8, N=L%16.
__global__ void __launch_bounds__(256)
k_mlp(const float* __restrict__ x,  const float* __restrict__ W1,
      const float* __restrict__ tw1, const float* __restrict__ b1,
      const float* __restrict__ W2,  const float* __restrict__ tw2,
      const float* __restrict__ b2,  const int* __restrict__ cnt,
      const int* __restrict__ off,   const int* __restrict__ perm,
      float* __restrict__ out) {
  const int e    = blockIdx.y;
  const int lane = threadIdx.x & 31;
  const int wave = threadIdx.x >> 5;
  const int half = lane >> 4;      // 0: lanes 0-15, 1: lanes 16-31
  const int nn   = lane & 15;      // position within half

  const int myCnt = cnt[e];
  const int myOff = off[e];
  const int tiles = (myCnt + 15) >> 4;
  if (tiles == 0) return;

  const float tw2e = tw2[e];
  const float b2e  = b2[e];

  // ---- preload invariant B fragments (W1[e]) into registers ----
  // 13 K-steps x 4 hidden tiles x 2 regs = 104 VGPRs, reused for every tile.
  const float* __restrict__ W1e = W1 + e * (INDIM * HIDD);
  float breg[13][4][2];
#pragma unroll
  for (int ks = 0; ks < 13; ++ks) {
    const int k0 = ks * 4 + 2 * half;
#pragma unroll
    for (int h = 0; h < 4; ++h) {
      const int col = h * 16 + nn;
      const bool cOK = (col < HIDD);
      breg[ks][h][0] = (cOK && (k0     < INDIM)) ? W1e[(k0)     * HIDD + col] : 0.0f;
      breg[ks][h][1] = (cOK && (k0 + 1 < INDIM)) ? W1e[(k0 + 1) * HIDD + col] : 0.0f;
    }
  }
  // per-lane epilogue weights for this lane's 4 hidden columns
  float tw1h[4], b1h[4], w2h[4];
#pragma unroll
  for (int h = 0; h < 4; ++h) {
    const int col = h * 16 + nn;
    const bool cOK = (col < HIDD);
    tw1h[h] = cOK ? tw1[e * HIDD + col] : 0.0f;
    b1h[h]  = cOK ? b1[e * HIDD + col]  : 0.0f;
    w2h[h]  = cOK ? W2[e * HIDD + col]  : 0.0f;
  }

  const int wavesTotal = gridDim.x * (blockDim.x >> 5);
  for (int tIdx = blockIdx.x * (blockDim.x >> 5) + wave; tIdx < tiles;
       tIdx += wavesTotal) {
    const int tBase = tIdx << 4;

    // This lane fetches sample M = nn (both halves fetch the same row).
    int mIdx = tBase + nn;
    if (mIdx > myCnt - 1) mIdx = myCnt - 1;          // clamp partial tile
    const int rowIdx = perm[myOff + mIdx];
    const float* __restrict__ xr = x + (size_t)rowIdx * ROWW;
    const float t = xr[0];

    // ---- A fragments: 13 x 2 regs of features (zero-pad K=50,51) ----
    float af[13][2];
#pragma unroll
    for (int ks = 0; ks < 13; ++ks) {
      const int k = ks * 4 + 2 * half;
      if (ks == 12 && half == 1) {
        af[ks][0] = 0.0f; af[ks][1] = 0.0f;
      } else {
        af[ks][0] = xr[1 + k];
        af[ks][1] = xr[2 + k];
      }
    }

    // ---- layer 1: h1(16x64) = A(16x50) x W1pad(50x64), f32 WMMA ----
    v8f acc[4];
#pragma unroll
    for (int h = 0; h < 4; ++h) acc[h] = (v8f){0,0,0,0,0,0,0,0};
#pragma unroll
    for (int ks = 0; ks < 13; ++ks) {
      v2f a; a.x = af[ks][0]; a.y = af[ks][1];
#pragma unroll
      for (int h = 0; h < 4; ++h) {
        v2f b; b.x = breg[ks][h][0]; b.y = breg[ks][h][1];
        acc[h] = __builtin_amdgcn_wmma_f32_16x16x4_f32(
            false, a, false, b, (short)0, acc[h], false, false);
      }
    }

    // ---- epilogue: bias + relu + dot(W2) + cross-lane reduce ----
    // acc[h][r] is h1[sample r+8*half][hid h*16+nn]
    float tr[8], part[8];
#pragma unroll
    for (int r = 0; r < 8; ++r) tr[r] = __shfl(t, r + 8 * half, 32);
#pragma unroll
    for (int r = 0; r < 8; ++r) part[r] = 0.0f;
#pragma unroll
    for (int h = 0; h < 4; ++h) {
#pragma unroll
      for (int r = 0; r < 8; ++r) {
        float v = acc[h][r] + tr[r] * tw1h[h] + b1h[h];
        v = fmaxf(v, 0.0f);
        part[r] = fmaf(v, w2h[h], part[r]);
      }
    }
    // reduce over the 16 hidden-lanes of each half (wave32: masks 1..8)
#pragma unroll
    for (int r = 0; r < 8; ++r) {
      part[r] += __shfl_xor(part[r], 1, 32);
      part[r] += __shfl_xor(part[r], 2, 32);
      part[r] += __shfl_xor(part[r], 4, 32);
      part[r] += __shfl_xor(part[r], 8, 32);
    }
    // ---- layer-2 bias and scattered store (sample m = r + 8*half) ----
#pragma unroll
    for (int r = 0; r < 8; ++r) {
      const int m = r + 8 * half;
      const float y = part[r] + tr[r] * tw2e + b2e;
      const int g = __shfl(rowIdx, m, 32);
      if (nn == r && (tBase + m) < myCnt) out[g] = y;
    }
  }
}

// ---------------------------------------------------------------------------
extern "C" void kernel_launch(void* const* d_in, const int* in_sizes, int n_in,
                              void* d_out, int out_size, void* d_ws, size_t ws_size,
                              hipStream_t stream) {
  const float* x   = (const float*)d_in[0];
  const float* W1  = (const float*)d_in[1];
  const float* tw1 = (const float*)d_in[2];
  const float* b1  = (const float*)d_in[3];
  const float* W2  = (const float*)d_in[4];
  const float* tw2 = (const float*)d_in[5];
  const float* b2  = (const float*)d_in[6];
  float* out = (float*)d_out;

  const int n = in_sizes[0] / ROWW;   // = NSAMP

  // workspace layout: cnt[8] | off[8] | cur[8] | perm[n]
  int* cnt  = (int*)d_ws;
  int* off  = cnt + 8;
  int* cur  = cnt + 16;
  int* perm = cnt + 24;

  k_init<<<1, 32, 0, stream>>>(cnt);
  k_count<<<512, 256, 0, stream>>>(x, cnt, n);
  k_prefix<<<1, 32, 0, stream>>>(cnt, off, cur);
  k_scatter<<<(n + 255) / 256, 256, 0, stream>>>(x, cur, perm, n);

  dim3 grid(160, NEXP);
  k_mlp<<<grid, 256, 0, stream>>>(x, W1, tw1, b1, W2, tw2, b2,
                                  cnt, off, perm, out);
}